// DistanceInfoNCE_88639535055182
// MI455X (gfx1250) — compile-verified
//
#include <hip/hip_runtime.h>
#include <hip/hip_bf16.h>
#include <math.h>

typedef __attribute__((ext_vector_type(2))) float v2f;
typedef __attribute__((ext_vector_type(8))) float v8f;

#define NROWS 4096
#define NDIM  256
#define TAU_F 0.5f
#define EPS_F 1e-9f

// K-split for occupancy: 4096/16 rows/wave alone gives only 256 waves on a
// machine that needs thousands of wave32s in flight to cover HBM latency at
// 23.3 TB/s. Splitting D=256 into 4 chunks of 64 gives 1024 waves (128
// blocks x 8 waves), each producing partial reductions combined later in a
// fixed (deterministic) order.
#define KSPLIT 4
#define KCHUNK (NDIM / KSPLIT)   // 64 columns per chunk

// ---------------------------------------------------------------------------
// K1a: partial column sums of anchor. 64 blocks x 256 threads.
// Block b owns rows [b*64, b*64+64). Thread d owns column d.
// Fully coalesced (consecutive threads -> consecutive addresses).
// ---------------------------------------------------------------------------
__global__ void __launch_bounds__(256)
k_colsum_partial(const float* __restrict__ A, float* __restrict__ partial) {
    const int b = blockIdx.x;
    const int d = threadIdx.x;
    const float* p = A + (size_t)(b * 64) * NDIM + d;
    float acc = 0.0f;
    #pragma unroll 4
    for (int r = 0; r < 64; ++r) acc += p[(size_t)r * NDIM];
    partial[b * NDIM + d] = acc;
}

// ---------------------------------------------------------------------------
// K1b: deterministic reduction of partials -> m[256]. 1 block x 256 threads.
// ---------------------------------------------------------------------------
__global__ void __launch_bounds__(256)
k_colsum_reduce(const float* __restrict__ partial, float* __restrict__ m) {
    const int d = threadIdx.x;
    float acc = 0.0f;
    #pragma unroll
    for (int b = 0; b < 64; ++b) acc += partial[b * NDIM + d];
    m[d] = acc;
}

// ---------------------------------------------------------------------------
// K2: per-row reductions via V_WMMA_F32_16X16X4_F32 with a K-split.
// Grid: 128 blocks x 256 threads. blockIdx.x = rowblk*KSPLIT + kchunk.
// Each block: 8 waves x 16 rows = 128 rows, columns [kchunk*64, kchunk*64+64).
//
// A-fragment layout (16x4 fp32, ISA 7.12.2): lane l holds row (l&15);
// VGPR0/VGPR1 hold K = klo, klo+1 where klo = (l>=16 ? 2 : 0).
// B-fragment (4x16): same K-to-(lane,vgpr) mapping; all our B columns are
// identical (broadcast m / ones) so only the K index matters per lane.
// D layout: VGPR j, lanes 0-15 -> row j; lanes 16-31 -> row 8+j. All N
// columns equal -> lane 0 carries rows 0..7, lane 16 carries rows 8..15,
// so extraction needs no cross-lane traffic.
// ---------------------------------------------------------------------------
__global__ void __launch_bounds__(256)
k_rows_wmma(const float* __restrict__ A, const float* __restrict__ P,
            const float* __restrict__ Ng, const float* __restrict__ m,
            float* __restrict__ sqp, float* __restrict__ posp,
            float* __restrict__ negp, float* __restrict__ dotp) {
    __shared__ float sm[KCHUNK];
    const int tid    = threadIdx.x;
    const int rowblk = blockIdx.x >> 2;        // 0..31
    const int kchunk = blockIdx.x & (KSPLIT - 1);
    const int k0     = kchunk * KCHUNK;

    if (tid < KCHUNK) sm[tid] = m[k0 + tid];
    __syncthreads();

    const int wave = tid >> 5;
    const int lane = tid & 31;
    const int rowbase = rowblk * 128 + wave * 16;
    const int r   = lane & 15;
    const int klo = (lane >> 4) << 1;   // 0 for lanes 0-15, 2 for lanes 16-31
    const int row = rowbase + r;

    const float* arow = A  + (size_t)row * NDIM + k0;
    const float* prow = P  + (size_t)row * NDIM + k0;
    const float* nrow = Ng + (size_t)row * NDIM + k0;

    v8f acc_sq  = {};
    v8f acc_pd  = {};
    v8f acc_nd  = {};
    v8f acc_dot = {};
    const v2f ones = {1.0f, 1.0f};

    for (int kk = 0; kk < KCHUNK; kk += 4) {
        float2 av = *(const float2*)(arow + kk + klo);
        float2 pv = *(const float2*)(prow + kk + klo);
        float2 nv = *(const float2*)(nrow + kk + klo);
        float2 mv = *(const float2*)(&sm[kk + klo]);   // ds_load_b64

        const float dpx = av.x - pv.x, dpy = av.y - pv.y;
        const float dnx = av.x - nv.x, dny = av.y - nv.y;

        v2f Aa  = {av.x, av.y};
        v2f Asq = {av.x * av.x, av.y * av.y};
        v2f Apd = {dpx * dpx, dpy * dpy};
        v2f And = {dnx * dnx, dny * dny};
        v2f Bm  = {mv.x, mv.y};

        acc_dot = __builtin_amdgcn_wmma_f32_16x16x4_f32(
            false, Aa,  false, Bm,   (short)0, acc_dot, false, false);
        acc_sq  = __builtin_amdgcn_wmma_f32_16x16x4_f32(
            false, Asq, false, ones, (short)0, acc_sq,  false, false);
        acc_pd  = __builtin_amdgcn_wmma_f32_16x16x4_f32(
            false, Apd, false, ones, (short)0, acc_pd,  false, false);
        acc_nd  = __builtin_amdgcn_wmma_f32_16x16x4_f32(
            false, And, false, ones, (short)0, acc_nd,  false, false);
    }

    // Lane 0 holds rows rowbase+0..7 in acc[0..7]; lane 16 holds rows +8..15.
    if (lane == 0 || lane == 16) {
        const int mrow = rowbase + ((lane >> 4) << 3);
        const int base = kchunk * NROWS + mrow;
        #pragma unroll
        for (int j = 0; j < 8; ++j) {
            sqp[base + j]  = acc_sq[j];
            posp[base + j] = acc_pd[j];
            negp[base + j] = acc_nd[j];
            dotp[base + j] = acc_dot[j];
        }
    }
}

// ---------------------------------------------------------------------------
// K3: finalize. 1 block x 256 threads, fixed-order reductions (deterministic).
// Combines the KSPLIT partials, then:
//   S[i]   = B*sq[i] + T - 2*(a_i . m) + neg_dist[i]
//   loss_i = -log(ps / (ps + ns + eps)),  out = mean(loss)
// ---------------------------------------------------------------------------
__global__ void __launch_bounds__(256)
k_finalize(const float* __restrict__ sqp, const float* __restrict__ posp,
           const float* __restrict__ negp, const float* __restrict__ dotp,
           float* __restrict__ out) {
    __shared__ float red[256];
    const int tid = threadIdx.x;

    float t = 0.0f;
    for (int i = tid; i < NROWS; i += 256) {
        float s = 0.0f;
        #pragma unroll
        for (int c = 0; c < KSPLIT; ++c) s += sqp[c * NROWS + i];
        t += s;
    }
    red[tid] = t;
    __syncthreads();
    for (int s = 128; s > 0; s >>= 1) {
        if (tid < s) red[tid] += red[tid + s];
        __syncthreads();
    }
    const float T = red[0];
    __syncthreads();

    float acc = 0.0f;
    for (int i = tid; i < NROWS; i += 256) {
        float sq_i = 0.0f, pd_i = 0.0f, nd_i = 0.0f, dt_i = 0.0f;
        #pragma unroll
        for (int c = 0; c < KSPLIT; ++c) {
            sq_i += sqp[c * NROWS + i];
            pd_i += posp[c * NROWS + i];
            nd_i += negp[c * NROWS + i];
            dt_i += dotp[c * NROWS + i];
        }
        const float S  = (float)NROWS * sq_i + T - 2.0f * dt_i + nd_i;
        const float ps = expf(-pd_i / TAU_F);
        const float ns = expf(-S / TAU_F);
        acc += -logf(ps / (ps + ns + EPS_F));
    }
    red[tid] = acc;
    __syncthreads();
    for (int s = 128; s > 0; s >>= 1) {
        if (tid < s) red[tid] += red[tid + s];
        __syncthreads();
    }
    if (tid == 0) out[0] = red[0] * (1.0f / (float)NROWS);
}

// ---------------------------------------------------------------------------
// Workspace layout (floats):
//   [0,256)          m        (column sums of anchor)
//   [256,16640)      partial  (64 x 256 column partials)
//   [16640,33024)    sqp      (KSPLIT x 4096)
//   [33024,49408)    posp
//   [49408,65792)    negp
//   [65792,82176)    dotp
// ~330 KB total; every element written before read on every call.
// ---------------------------------------------------------------------------
extern "C" void kernel_launch(void* const* d_in, const int* in_sizes, int n_in,
                              void* d_out, int out_size, void* d_ws, size_t ws_size,
                              hipStream_t stream) {
    (void)in_sizes; (void)n_in; (void)out_size; (void)ws_size;
    const float* A  = (const float*)d_in[0];
    const float* P  = (const float*)d_in[1];
    const float* Ng = (const float*)d_in[2];

    float* ws      = (float*)d_ws;
    float* m       = ws;
    float* partial = ws + 256;
    float* sqp     = ws + 256 + 64 * 256;
    float* posp    = sqp  + KSPLIT * NROWS;
    float* negp    = posp + KSPLIT * NROWS;
    float* dotp    = negp + KSPLIT * NROWS;

    k_colsum_partial<<<64,  256, 0, stream>>>(A, partial);
    k_colsum_reduce <<<1,   256, 0, stream>>>(partial, m);
    k_rows_wmma     <<<128, 256, 0, stream>>>(A, P, Ng, m, sqp, posp, negp, dotp);
    k_finalize      <<<1,   256, 0, stream>>>(sqp, posp, negp, dotp, (float*)d_out);
}